// TersoffAttention_12128987644521
// MI455X (gfx1250) — compile-verified
//
#include <hip/hip_runtime.h>
#include <hip/hip_bf16.h>
#include <stdint.h>

// TersoffAttention on MI455X (gfx1250).
// E=100000 edges, T=1000000 triplets, C=64 channels.
// Design: L2-resident gather/scatter (xij + ft = 51MB << 192MB L2);
// - batched triplet-cosine dots via V_WMMA_F32_16X16X4_F32 (diag of A*B),
// - wave-per-triplet logits, 2 channels/lane (wave32), async-to-LDS
//   double-buffered row gathers (ASYNCcnt pipeline),
// - ordered-uint atomic max + f32 atomic adds for segment softmax,
// - global_prefetch_b8 ahead of the aggregation gathers.
// Workspace: (4E + 3T + 2E) floats ~= 14.4 MB.

typedef __attribute__((ext_vector_type(2))) float v2f;
typedef __attribute__((ext_vector_type(8))) float v8f;

__device__ __forceinline__ unsigned f32_order_enc(float f) {
  unsigned u = __float_as_uint(f);
  return (u & 0x80000000u) ? ~u : (u | 0x80000000u);
}
__device__ __forceinline__ float f32_order_dec(unsigned k) {
  return __uint_as_float((k & 0x80000000u) ? (k & 0x7fffffffu) : ~k);
}

// ---- Phase 0: per-edge unit bond directions (negated) + init softmax stats
__global__ __launch_bounds__(256) void k_init_edges(
    const float* __restrict__ r, float* __restrict__ rnorm4,
    unsigned* __restrict__ amax, float* __restrict__ denom, int E) {
  int i = blockIdx.x * blockDim.x + threadIdx.x;
  if (i >= E) return;
  float rx = r[3 * i + 0], ry = r[3 * i + 1], rz = r[3 * i + 2];
  float inv = rsqrtf(rx * rx + ry * ry + rz * rz);
  rnorm4[4 * i + 0] = -rx * inv;
  rnorm4[4 * i + 1] = -ry * inv;
  rnorm4[4 * i + 2] = -rz * inv;
  rnorm4[4 * i + 3] = 0.f;
  amax[i] = 0u;     // ordered-encoding floor (below all real keys)
  denom[i] = 0.f;
}

__global__ __launch_bounds__(256) void k_zero4(float4* __restrict__ p, int n4) {
  int i = blockIdx.x * blockDim.x + threadIdx.x;
  int stride = gridDim.x * blockDim.x;
  for (; i < n4; i += stride) p[i] = make_float4(0.f, 0.f, 0.f, 0.f);
}

// ---- Phase A0: 16 triplet-angle thetas per wave via one f32 WMMA.
// A = 16x4 src vectors, B = 4x16 dst vectors; diag(A*B) = cos_jik.
// 16x4 A layout (ISA 7.12.2): lane L holds row M=L%16; low lane-half holds
// K={0,1} in vgpr{0,1}, high half K={2,3}. B (4x16) is the mirror with
// N=L%16. So each lane packs (v.x,v.y) or (v.z,0).
__global__ __launch_bounds__(256) void k_triplet_theta_wmma(
    const float* __restrict__ rnorm4, const int* __restrict__ t_src,
    const int* __restrict__ t_dst, float* __restrict__ theta_t, int T) {
  const int lane = threadIdx.x & 31;
  const int wave = (blockIdx.x * blockDim.x + threadIdx.x) >> 5;
  const int tb = wave * 16;
  if (tb >= T) return;                       // uniform per wave
  const int m = lane & 15;
  const bool hi = lane >= 16;
  const int tt = tb + m;
  const int tc = (tt < T) ? tt : 0;          // clamp; store guarded below
  const int is = t_src[tc], id = t_dst[tc];
  const float4 vs = *(const float4*)(rnorm4 + 4 * (size_t)is);
  const float4 vd = *(const float4*)(rnorm4 + 4 * (size_t)id);
  v2f a, b;
  a.x = hi ? vs.z : vs.x;  a.y = hi ? 0.f : vs.y;
  b.x = hi ? vd.z : vd.x;  b.y = hi ? 0.f : vd.y;
  v8f acc = {0.f, 0.f, 0.f, 0.f, 0.f, 0.f, 0.f, 0.f};
  acc = __builtin_amdgcn_wmma_f32_16x16x4_f32(
      /*neg_a=*/false, a, /*neg_b=*/false, b,
      /*c_mod=*/(short)0, acc, /*reuse_a=*/false, /*reuse_b=*/false);
  // Extract diagonal D[n][n]: n<8 -> vgpr n @ lane n; n>=8 -> vgpr n-8 @ lane n+16.
  float val = acc[lane & 7];
  float dg = __shfl(val, (lane < 8) ? lane : lane + 16, 32);
  if (lane < 16 && tt < T) {
    float x = fminf(fmaxf(dg, -1.f + 1e-6f), 1.f - 1e-6f);
    theta_t[tt] = acosf(x);
  }
}

// ---- Phase A: per-triplet logit a[t] + segment max (wave per triplet)
// LDS: 8 waves * 2 buffers * (64 src + 64 dst floats) = 8 KB / block.
__global__ __launch_bounds__(256) void k_triplet_logits(
    const float* __restrict__ xij, const float* __restrict__ theta_t,
    const float* __restrict__ attn, const int* __restrict__ t_src,
    const int* __restrict__ t_dst, float* __restrict__ a_t,
    unsigned* __restrict__ amax, int T) {
  __shared__ float shm[8 * 2 * 128];
  const int lane = threadIdx.x & 31;
  const int wIb = threadIdx.x >> 5;
  const int W = (gridDim.x * blockDim.x) >> 5;      // total waves
  const int wave = (blockIdx.x * blockDim.x + threadIdx.x) >> 5;

  float* wbase = &shm[wIb * 256];
  // LDS byte address for this lane's 16B slot (generic->LDS: low 32 bits)
  const unsigned lds0 = (unsigned)(uintptr_t)wbase + (unsigned)(lane * 16);
  const float2 at2 = *(const float2*)(attn + 2 * lane);

  // Issue one async b128/lane: lanes 0..15 fetch src row, 16..31 dst row.
  auto issue = [&](int t, int buf) {
    int s = t_src[t], d = t_dst[t];
    const float* row = (lane < 16) ? (xij + (size_t)s * 64)
                                   : (xij + (size_t)d * 64);
    unsigned long long ga =
        (unsigned long long)(uintptr_t)row + (unsigned long long)((lane & 15) * 16);
    unsigned la = lds0 + (unsigned)(buf * 512);
    asm volatile("global_load_async_to_lds_b128 %0, %1, off"
                 :: "v"(la), "v"(ga) : "memory");
  };

  int t = wave;
  int buf = 0;
  if (t < T) issue(t, 0);
  for (; t < T; t += W) {
    int tn = t + W;
    if (tn < T) {
      issue(tn, buf ^ 1);
      asm volatile("s_wait_asynccnt 0x1" ::: "memory");  // iter t's copy done
    } else {
      asm volatile("s_wait_asynccnt 0x0" ::: "memory");
    }

    const int d = t_dst[t];
    const float theta = theta_t[t];                    // uniform bcast load

    const float* b = wbase + buf * 128;
    float2 xs = *(const float2*)(b + 2 * lane);        // xij[src][c0,c1]
    float2 xd = *(const float2*)(b + 64 + 2 * lane);   // xij[dst][c0,c1]

    float c0 = (float)(2 * lane), c1 = (float)(2 * lane + 1);
    float v0 = __cosf(theta * c0) + xs.x + xd.x;       // z + x_jik
    float v1 = __cosf(theta * c1) + xs.y + xd.y;
    float e0 = v0 / (1.f + __expf(-v0));               // silu
    float e1 = v1 / (1.f + __expf(-v1));
    float p = e0 * at2.x + e1 * at2.y;

    #pragma unroll
    for (int o = 16; o > 0; o >>= 1) p += __shfl_xor(p, o, 32);

    if (lane == 0) {
      a_t[t] = p;
      atomicMax(&amax[d], f32_order_enc(p));
    }
    buf ^= 1;
  }
}

// ---- Phase B: exp(a - max[dst]) and segment denom (thread per triplet)
__global__ __launch_bounds__(256) void k_softmax_norm(
    const int* __restrict__ t_dst, const float* __restrict__ a_t,
    const unsigned* __restrict__ amax, float* __restrict__ ex_t,
    float* __restrict__ denom, int T) {
  int i = blockIdx.x * blockDim.x + threadIdx.x;
  if (i >= T) return;
  int d = t_dst[i];
  float e = __expf(a_t[i] - f32_order_dec(amax[d]));
  ex_t[i] = e;
  atomicAdd(&denom[d], e);
}

// ---- Phase C: ft[dst] += xij[src] * alpha (wave per triplet, atomics in L2)
__global__ __launch_bounds__(256) void k_aggregate(
    const float* __restrict__ xij, const int* __restrict__ t_src,
    const int* __restrict__ t_dst, const float* __restrict__ ex_t,
    const float* __restrict__ denom, float* __restrict__ ft, int T) {
  const int lane = threadIdx.x & 31;
  const int W = (gridDim.x * blockDim.x) >> 5;
  int wave = (blockIdx.x * blockDim.x + threadIdx.x) >> 5;
  for (int t = wave; t < T; t += W) {
    int tn = t + W;
    if (tn < T) {   // prefetch next src row -> global_prefetch_b8
      int sn = t_src[tn];
      __builtin_prefetch(xij + (size_t)sn * 64 + 2 * lane, 0, 0);
    }
    int s = t_src[t], d = t_dst[t];
    float alpha = ex_t[t] / denom[d];
    float2 xv = *(const float2*)(xij + (size_t)s * 64 + 2 * lane);
    float* o = ft + (size_t)d * 64 + 2 * lane;
    atomicAdd(o + 0, xv.x * alpha);
    atomicAdd(o + 1, xv.y * alpha);
  }
}

extern "C" void kernel_launch(void* const* d_in, const int* in_sizes, int n_in,
                              void* d_out, int out_size, void* d_ws, size_t ws_size,
                              hipStream_t stream) {
  const float* xij  = (const float*)d_in[0];
  const float* r    = (const float*)d_in[1];
  const float* attn = (const float*)d_in[2];
  const int*   t_src = (const int*)d_in[3];
  const int*   t_dst = (const int*)d_in[4];
  const int E = in_sizes[1] / 3;
  const int C = in_sizes[2];        // 64 (kernels assume C==64: 2 ch/lane)
  const int T = in_sizes[3];
  float* ft = (float*)d_out;

  char* ws = (char*)d_ws;
  float* rnorm4   = (float*)ws;  ws += (size_t)E * 4 * sizeof(float);
  float* theta_t  = (float*)ws;  ws += (size_t)T * sizeof(float);
  float* a_t      = (float*)ws;  ws += (size_t)T * sizeof(float);
  float* ex_t     = (float*)ws;  ws += (size_t)T * sizeof(float);
  unsigned* amax  = (unsigned*)ws; ws += (size_t)E * sizeof(unsigned);
  float* denom    = (float*)ws;

  k_init_edges<<<(E + 255) / 256, 256, 0, stream>>>(r, rnorm4, amax, denom, E);
  k_zero4<<<2048, 256, 0, stream>>>((float4*)ft, (E * C) / 4);

  const long long waves16 = ((long long)T + 15) / 16;
  const int blocksTheta = (int)((waves16 * 32 + 255) / 256);
  k_triplet_theta_wmma<<<blocksTheta, 256, 0, stream>>>(rnorm4, t_src, t_dst,
                                                        theta_t, T);

  k_triplet_logits<<<8192, 256, 0, stream>>>(xij, theta_t, attn, t_src, t_dst,
                                             a_t, amax, T);
  k_softmax_norm<<<(T + 255) / 256, 256, 0, stream>>>(t_dst, a_t, amax, ex_t,
                                                      denom, T);
  k_aggregate<<<8192, 256, 0, stream>>>(xij, t_src, t_dst, ex_t, denom, ft, T);
}